// RepulsionEnergyLearnedRadius_2628519985581
// MI455X (gfx1250) — compile-verified
//
#include <hip/hip_runtime.h>
#include <hip/hip_bf16.h>
#include <math.h>

// ---- problem constants (match reference) ----
#define NUM_AA   20
#define EMB_DIM  16
#define KNN      64
#define EXCL     2
#define R_ON     8.0f
#define R_CUTF   10.0f
#define R_CUT2   100.0f
#define DELTA    0.3f
#define WALL     10.0f
#define RHO_MIN  1.6f
#define RHO_MAX  2.8f
#define BATCH    8
#define LSEQ     2048

// ---- kernel tiling ----
#define ROWS        16          // rows per block (one WMMA M-tile)
#define NWAVE       16
#define BLOCK_THR   (NWAVE * 32)
#define NCOLTILE    (LSEQ / 16) // 128 column tiles of 16
#define CAP         512         // per-row candidate capacity = 32 lanes * 16 slots
#define PERLANE     (CAP / 32)  // 16 records per lane for the bisection

typedef __attribute__((ext_vector_type(2))) float v2f;
typedef __attribute__((ext_vector_type(8))) float v8f;

__device__ __forceinline__ float warp_sum_f(float v) {
    #pragma unroll
    for (int o = 16; o > 0; o >>= 1) v += __shfl_xor(v, o, 32);
    return v;
}
__device__ __forceinline__ int warp_sum_i(int v) {
    #pragma unroll
    for (int o = 16; o > 0; o >>= 1) v += __shfl_xor(v, o, 32);
    return v;
}

__global__ void __launch_bounds__(BLOCK_THR)
repulsion_knn_wmma_kernel(const float* __restrict__ R,
                          const int*   __restrict__ seq,
                          const float* __restrict__ emb,
                          const float* __restrict__ size_w,
                          const float* __restrict__ size_b,
                          float*       __restrict__ out)
{
    __shared__ float         rhoTab[32];          // learned radius per AA id
    __shared__ float         colNorm[LSEQ];       // |r_j|^2 for all columns
    __shared__ unsigned char colSeq[LSEQ];        // AA id per column
    __shared__ float         rowNorm[ROWS];
    __shared__ float         rowRho[ROWS];
    __shared__ unsigned int  cnt[ROWS];
    __shared__ alignas(16) float         dlist[ROWS * CAP]; // candidate d^2
    __shared__ alignas(16) unsigned char slist[ROWS * CAP]; // candidate AA ids

    const int tid  = threadIdx.x;
    const int lane = tid & 31;
    const int wid  = tid >> 5;
    const int blk  = blockIdx.x;
    const int b    = blk / (LSEQ / ROWS);
    const int rowBase = (blk % (LSEQ / ROWS)) * ROWS;
    const size_t batchOff = (size_t)b * LSEQ;

    // ---- Phase 0: rho table (20 entries), counters ----
    if (tid < NUM_AA) {
        float x = size_b[0];
        #pragma unroll
        for (int d = 0; d < EMB_DIM; ++d) x += emb[tid * EMB_DIM + d] * size_w[d];
        float s = 1.0f / (1.0f + expf(-x));
        rhoTab[tid] = RHO_MIN + (RHO_MAX - RHO_MIN) * s;
    }
    if (tid < ROWS) cnt[tid] = 0u;
    __syncthreads();

    // column norms + AA ids for whole sequence of this batch
    for (int c = tid; c < LSEQ; c += BLOCK_THR) {
        const float* p = R + (batchOff + c) * 3;
        float x = p[0], y = p[1], z = p[2];
        colNorm[c] = x * x + y * y + z * z;
        colSeq[c]  = (unsigned char)seq[batchOff + c];
    }
    if (tid < ROWS) {
        int gi = rowBase + tid;
        const float* p = R + (batchOff + gi) * 3;
        float x = p[0], y = p[1], z = p[2];
        rowNorm[tid] = x * x + y * y + z * z;
        rowRho[tid]  = rhoTab[(int)(unsigned char)seq[batchOff + gi]];
    }
    __syncthreads();

    // ---- Phase 1: Gram tiles via V_WMMA_F32_16X16X4_F32, filter d^2 < R_CUT^2 ----
    // A (16x4, f32): lanes 0-15 hold M=lane, K=0 (v0) / K=1 (v1);
    //                lanes 16-31 hold M=lane-16, K=2 (v0) / K=3 (v1). K=3 padded 0.
    const int m16  = lane & 15;
    const bool loHalf = (lane < 16);
    const unsigned laneBit    = 1u << lane;
    const unsigned prefixMask = laneBit - 1u;
    const unsigned halfSelect = loHalf ? 0x0000FFFFu : 0xFFFF0000u;

    const int arow = rowBase + m16;
    const float* ap = R + (batchOff + arow) * 3;
    float ax = ap[0], ay = ap[1], az = ap[2];
    v2f a;
    a.x = loHalf ? ax : az;            // x | z
    a.y = loHalf ? ay : 0.0f;          // y | 0 (pad)

    // preload row norms for my half's 8 rows into registers
    float rn[8];
    #pragma unroll
    for (int g = 0; g < 8; ++g) rn[g] = rowNorm[g + (loHalf ? 0 : 8)];

    for (int t = wid; t < NCOLTILE; t += NWAVE) {
        const int col = t * 16 + m16;
        const float* bp = R + (batchOff + col) * 3;
        float bx = bp[0], by = bp[1], bz = bp[2];
        v2f bb;
        bb.x = loHalf ? bx : bz;
        bb.y = loHalf ? by : 0.0f;

        v8f c = {};
        c = __builtin_amdgcn_wmma_f32_16x16x4_f32(
                /*neg_a=*/false, a, /*neg_b=*/false, bb,
                /*c_mod=*/(short)0, c, /*reuse_a=*/false, /*reuse_b=*/false);

        const float cn = colNorm[col];
        const unsigned char cs = colSeq[col];

        #pragma unroll
        for (int g = 0; g < 8; ++g) {
            const int mm = g + (loHalf ? 0 : 8);     // row within tile (per half-wave)
            const int gi = rowBase + mm;             // global row index
            int dd = gi - col; dd = dd < 0 ? -dd : dd;

            // squared distance only — sqrt deferred to Phase 2 survivors
            float d2 = fmaxf(rn[g] + cn - 2.0f * c[g], 1e-12f);
            const bool pred = (dd > EXCL) && (d2 < R_CUT2);

            // wave-wide ballot; the two 16-lane halves are two different rows
            unsigned mask = __builtin_amdgcn_ballot_w32(pred);
            unsigned halfMask = mask & halfSelect;
            unsigned hc = (unsigned)__popc(halfMask);

            // one LDS atomic per half-wave leader (lanes 0 / 16), unconditional
            unsigned base = 0;
            if (lane == 0 || lane == 16) base = atomicAdd(&cnt[mm], hc);
            // broadcast via readlane (no ds_bpermute)
            unsigned b0  = (unsigned)__builtin_amdgcn_readlane((int)base, 0);
            unsigned b16 = (unsigned)__builtin_amdgcn_readlane((int)base, 16);
            base = loHalf ? b0 : b16;

            if (pred) {
                unsigned rank = (unsigned)__popc(mask & prefixMask & halfSelect);
                unsigned pos = base + rank;
                if (pos < CAP) {
                    dlist[mm * CAP + pos] = d2;
                    slist[mm * CAP + pos] = cs;
                }
            }
        }
    }
    __syncthreads();

    // ---- Phase 2: one wave per row ----
    const int m = wid;                       // NWAVE == ROWS
    const int n = (int)min(cnt[m], (unsigned)CAP);

    // register-resident copy of this lane's contiguous 16-record chunk
    // (4x ds_load_b128) — used only for the bisection compares
    float d2r[PERLANE];
    {
        const float4* p4 = (const float4*)(dlist + m * CAP + lane * PERLANE);
        float4 q0 = p4[0], q1 = p4[1], q2 = p4[2], q3 = p4[3];
        d2r[0]=q0.x;  d2r[1]=q0.y;  d2r[2]=q0.z;  d2r[3]=q0.w;
        d2r[4]=q1.x;  d2r[5]=q1.y;  d2r[6]=q1.z;  d2r[7]=q1.w;
        d2r[8]=q2.x;  d2r[9]=q2.y;  d2r[10]=q2.z; d2r[11]=q2.w;
        d2r[12]=q3.x; d2r[13]=q3.y; d2r[14]=q3.z; d2r[15]=q3.w;
    }
    // pad unwritten slots so poison/garbage can never be counted
    const int laneBase = lane * PERLANE;
    #pragma unroll
    for (int j = 0; j < PERLANE; ++j)
        if (laneBase + j >= n) d2r[j] = 3.0e38f;

    float thr2 = R_CUT2;                     // keep everything if n <= KNN
    if (n > KNN) {
        // positive-float ordering == unsigned-int ordering of the bit patterns
        unsigned lo = 0u, hi = __float_as_uint(R_CUT2);
        #pragma unroll 1
        while (hi - lo > 1u) {
            unsigned midBits = lo + ((hi - lo) >> 1);
            float mid = __uint_as_float(midBits);
            int c = 0;
            #pragma unroll
            for (int j = 0; j < PERLANE; ++j) c += (d2r[j] <= mid) ? 1 : 0;
            c = warp_sum_i(c);
            if (c == KNN) { lo = midBits; break; }     // exact 64th-smallest
            if (c > KNN) hi = midBits; else lo = midBits;
        }
        thr2 = __uint_as_float(lo);          // count(d2 <= thr2) <= 64
    }

    // rolled energy loop from LDS: ONE copy of the transcendental body,
    // executed (under divergent exec) only for the <=64 survivors
    const float* dl = dlist + m * CAP;
    const unsigned char* sl = slist + m * CAP;
    const float rho_i = rowRho[m];
    const float invDelta = 1.0f / (DELTA + 1e-12f);
    const float invWin = 1.0f / (R_CUTF - R_ON);
    float e = 0.0f;
    #pragma unroll 1
    for (int i = lane; i < n; i += 32) {
        float d2 = dl[i];
        if (d2 <= thr2) {
            float d  = sqrtf(d2);            // d2 already clamped >= 1e-12
            float r0 = rho_i + rhoTab[(int)sl[i]];
            float x  = (r0 - d) * invDelta;
            // numerically stable softplus
            float sp = (x > 0.0f) ? (x + log1pf(expf(-x))) : log1pf(expf(x));
            float tt = fminf(fmaxf((d - R_ON) * invWin, 0.0f), 1.0f);
            float sw = 1.0f - tt * tt * (3.0f - 2.0f * tt);
            e += WALL * sp * sw;
        }
    }
    e = warp_sum_f(e);
    if (lane == 0) atomicAdd(&out[b], e);
}

extern "C" void kernel_launch(void* const* d_in, const int* in_sizes, int n_in,
                              void* d_out, int out_size, void* d_ws, size_t ws_size,
                              hipStream_t stream) {
    const float* R      = (const float*)d_in[0];
    const int*   seq    = (const int*)d_in[1];
    const float* emb    = (const float*)d_in[2];
    const float* size_w = (const float*)d_in[3];
    const float* size_b = (const float*)d_in[4];
    float* out = (float*)d_out;

    hipMemsetAsync(out, 0, (size_t)out_size * sizeof(float), stream);

    dim3 grid(BATCH * (LSEQ / ROWS));   // 8 * 128 = 1024 blocks
    dim3 block(BLOCK_THR);              // 16 wave32 waves
    repulsion_knn_wmma_kernel<<<grid, block, 0, stream>>>(R, seq, emb, size_w, size_b, out);
}